// ImprovedLoss_2783138807899
// MI455X (gfx1250) — compile-verified
//
#include <hip/hip_runtime.h>
#include <hip/hip_bf16.h>

#define BATCH    32
#define ANCH     75600      // 25200*27/9 anchors per sample
#define NF       9          // 4 box + 1 obj + 4 cls
#define NC       4
#define KTOP     300
#define NT       50
#define SORTN    512
#define THREADS  256
#define BETA     0.1f
#define SMOOTH_POS 0.95f
#define SMOOTH_NEG (0.05f / 3.0f)

typedef __attribute__((ext_vector_type(2))) float v2f;
typedef __attribute__((ext_vector_type(8))) float v8f;

__device__ __forceinline__ float softplusf(float x) {
    return fmaxf(x, 0.0f) + log1pf(expf(-fabsf(x)));
}
__device__ __forceinline__ float sigmoidf(float x) {
    return 1.0f / (1.0f + expf(-x));
}
// order-preserving float -> uint key (larger float => larger key)
__device__ __forceinline__ unsigned fkey(float x) {
    unsigned u = __float_as_uint(x);
    return (u & 0x80000000u) ? ~u : (u | 0x80000000u);
}

__global__ __launch_bounds__(THREADS) void yolo_loss_main(
    const float* __restrict__ pred, const float* __restrict__ tboxg,
    const int* __restrict__ tclsg, float* __restrict__ ws)
{
    __shared__ unsigned s_hist[256];
    __shared__ unsigned long long s_key[SORTN];   // (~u)<<32 | idx ; ascending sort
    __shared__ float s_part[THREADS];
    __shared__ float4 s_bx[KTOP];
    __shared__ float4 s_tb[NT];
    __shared__ int   s_tc[NT];
    __shared__ int   s_chosen[NT];
    __shared__ float s_cls[NT], s_box[NT], s_objc[NT];
    __shared__ int   s_last[NT];
    __shared__ unsigned s_prefix, s_need, s_cgt, s_ceq;
    __shared__ float s_spsum;

    const int b   = blockIdx.x;
    const int tid = threadIdx.x;
    const float* p = pred + (size_t)b * ANCH * NF;

    // ---------------- pass 0: softplus sum + MSB-byte histogram ----------------
    if (tid < 256) s_hist[tid] = 0u;
    if (tid == 0) { s_need = KTOP; s_prefix = 0u; }
    __syncthreads();

    float sp = 0.0f;
    for (int i = tid; i < ANCH; i += THREADS) {
        const float* q = p + (size_t)i * NF;
        __builtin_prefetch(q + (size_t)THREADS * NF, 0, 0);   // global_prefetch_b8
        float x = q[4];
        sp += softplusf(x);
        unsigned u = fkey(x);
        atomicAdd(&s_hist[u >> 24], 1u);
    }
    s_part[tid] = sp;
    __syncthreads();

    // ---- block reduction of 256 partials via chained f32 WMMA (D = A x ones) ----
    if ((tid >> 5) == 0) {            // wave 0 only: wave-uniform branch, EXEC all ones
        int lane = tid;
        int m  = lane & 15;
        int kb = (lane < 16) ? 0 : 2; // 16x4 f32 A layout: v0:K=0/2, v1:K=1/3
        float total;
#if defined(__has_builtin) && __has_builtin(__builtin_amdgcn_wmma_f32_16x16x4_f32)
        v8f acc = {};
        v2f ones; ones.x = 1.0f; ones.y = 1.0f;
#pragma unroll
        for (int q = 0; q < 4; ++q) {  // 4 chunks of 64 partials
            v2f aa;
            aa.x = s_part[q * 64 + m * 4 + kb];
            aa.y = s_part[q * 64 + m * 4 + kb + 1];
            acc = __builtin_amdgcn_wmma_f32_16x16x4_f32(
                false, aa, false, ones, (short)0, acc, false, false);
        }
        // D[m][n] = rowsum(A[m]); col N=0 lives on lane 0 (M=0..7) and lane 16 (M=8..15)
        float srow = acc[0] + acc[1] + acc[2] + acc[3] +
                     acc[4] + acc[5] + acc[6] + acc[7];
        total = __shfl(srow, 0, 32) + __shfl(srow, 16, 32);
#else
        float v = 0.0f;
        for (int q = lane; q < THREADS; q += 32) v += s_part[q];
        for (int off = 16; off > 0; off >>= 1) v += __shfl_down(v, off, 32);
        total = __shfl(v, 0, 32);
#endif
        if (lane == 0) s_spsum = total;
    }
    __syncthreads();

    // ---------------- radix top-k select: byte levels MSB -> LSB ----------------
    for (int level = 0; level < 4; ++level) {
        const int shift = 24 - level * 8;
        if (level > 0) {
            if (tid < 256) s_hist[tid] = 0u;
            __syncthreads();
            const unsigned pfx   = s_prefix;
            const unsigned hmask = 0xFFFFFFFFu << (shift + 8);
            for (int i = tid; i < ANCH; i += THREADS) {
                unsigned u = fkey(p[(size_t)i * NF + 4]);
                if ((u & hmask) == pfx) atomicAdd(&s_hist[(u >> shift) & 0xFFu], 1u);
            }
            __syncthreads();
        }
        if (tid == 0) {
            unsigned need = s_need, cum = 0u; int sel = 0;
            for (int bin = 255; bin >= 0; --bin) {
                unsigned c = s_hist[bin];
                if (cum + c >= need) { sel = bin; s_need = need - cum; break; }
                cum += c;
            }
            s_prefix |= ((unsigned)sel) << shift;
        }
        __syncthreads();
    }

    const unsigned T       = s_prefix;           // k-th largest key
    const unsigned take_eq = s_need;             // how many ties at T to keep
    const unsigned n_gt    = KTOP - take_eq;
    if (tid == 0) { s_cgt = 0u; s_ceq = 0u; }
    __syncthreads();

    // ---------------- gather top-k (value,index) pairs ----------------
    for (int i = tid; i < ANCH; i += THREADS) {
        unsigned u = fkey(p[(size_t)i * NF + 4]);
        if (u > T) {
            unsigned pos = atomicAdd(&s_cgt, 1u);
            s_key[pos] = (((unsigned long long)(~u)) << 32) | (unsigned)i;
        } else if (u == T) {
            unsigned pos = atomicAdd(&s_ceq, 1u);
            if (pos < take_eq)
                s_key[n_gt + pos] = (((unsigned long long)(~u)) << 32) | (unsigned)i;
        }
    }
    for (int i = KTOP + tid; i < SORTN; i += THREADS)
        s_key[i] = 0xFFFFFFFFFFFFFFFFULL;        // pad sorts last
    __syncthreads();

    // ---------------- bitonic sort (ascending by (~val, idx)) ----------------
    for (int k = 2; k <= SORTN; k <<= 1) {
        for (int j = k >> 1; j > 0; j >>= 1) {
            for (int t = tid; t < SORTN; t += THREADS) {
                int partner = t ^ j;
                if (partner > t) {
                    bool up = ((t & k) == 0);
                    unsigned long long a = s_key[t], c = s_key[partner];
                    if ((a > c) == up) { s_key[t] = c; s_key[partner] = a; }
                }
            }
            __syncthreads();
        }
    }

    // ---------------- decode top-k boxes, load targets ----------------
    for (int j = tid; j < KTOP; j += THREADS) {
        int idx = (int)(s_key[j] & 0xFFFFFFFFu);
        const float* q = p + (size_t)idx * NF;
        float4 bx;
        bx.x = sigmoidf(q[0]); bx.y = sigmoidf(q[1]);
        bx.z = sigmoidf(q[2]); bx.w = sigmoidf(q[3]);
        s_bx[j] = bx;
    }
    for (int t = tid; t < NT; t += THREADS) {
        const float* tq = tboxg + ((size_t)b * NT + t) * 4;
        float4 tb;
        tb.x = fminf(fmaxf(tq[0], 0.f), 1.f);
        tb.y = fminf(fmaxf(tq[1], 0.f), 1.f);
        tb.z = fminf(fmaxf(tq[2], 0.f), 1.f);
        tb.w = fminf(fmaxf(tq[3], 0.f), 1.f);
        s_tb[t] = tb;
        int c = tclsg[b * NT + t];
        s_tc[t] = min(max(c, 0), NC - 1);
    }
    __syncthreads();

    // ---------------- IoU + argmax (first-max keeps topk order semantics) ----------------
    if (tid < NT) {
        float4 tb = s_tb[tid];
        float thw = tb.z * 0.5f, thh = tb.w * 0.5f;
        float tx1 = tb.x - thw, ty1 = tb.y - thh, tx2 = tb.x + thw, ty2 = tb.y + thh;
        float area2 = (tx2 - tx1) * (ty2 - ty1);
        float best = -1.0f; int bestj = 0;
        for (int j = 0; j < KTOP; ++j) {
            float4 pb = s_bx[j];
            float hw = pb.z * 0.5f, hh = pb.w * 0.5f;
            float px1 = pb.x - hw, py1 = pb.y - hh, px2 = pb.x + hw, py2 = pb.y + hh;
            float iw = fmaxf(fminf(px2, tx2) - fmaxf(px1, tx1), 0.f);
            float ih = fmaxf(fminf(py2, ty2) - fmaxf(py1, ty1), 0.f);
            float inter = iw * ih;
            float area1 = (px2 - px1) * (py2 - py1);
            float uni = area1 + area2 - inter;
            float iou = (uni > 0.f) ? inter / uni : 0.f;
            if (iou > best) { best = iou; bestj = j; }
        }
        s_chosen[tid] = (int)(s_key[bestj] & 0xFFFFFFFFu);
    }
    __syncthreads();

    // ---------------- per-target contributions (dedup: last occurrence wins) ----------------
    if (tid < NT) {
        int a = s_chosen[tid];
        bool lastOcc = true;
        for (int t2 = tid + 1; t2 < NT; ++t2)
            if (s_chosen[t2] == a) { lastOcc = false; break; }
        float clsv = 0.f, boxv = 0.f, objv = 0.f;
        if (lastOcc) {
            unsigned mask = 0u;                     // all (a, tcls) pairs get smooth_pos
            for (int t2 = 0; t2 < NT; ++t2)
                if (s_chosen[t2] == a) mask |= 1u << s_tc[t2];
            const float* q = p + (size_t)a * NF;
            for (int c = 0; c < NC; ++c) {
                float y = ((mask >> c) & 1u) ? SMOOTH_POS : SMOOTH_NEG;
                float x = q[5 + c];
                clsv += softplusf(x) - x * y;
            }
            float xo = q[4];
            objv = 1.5f * softplusf(xo) - 2.0f * xo;   // correction vs 0.5*softplus baseline
            float4 tb = s_tb[tid];
            float tgt[4] = { tb.x, tb.y, tb.z, tb.w };
            for (int d = 0; d < 4; ++d) {
                float bp = sigmoidf(q[d]);
                float df = fabsf(bp - tgt[d]);
                boxv += (df < BETA) ? (0.5f * df * df / BETA) : (df - 0.5f * BETA);
            }
        }
        s_cls[tid] = clsv; s_box[tid] = boxv; s_objc[tid] = objv;
        s_last[tid] = lastOcc ? 1 : 0;
    }
    __syncthreads();

    if (tid == 0) {
        float cls_sum = 0.f, box_sum = 0.f, obj_corr = 0.f; int npos = 0;
        for (int t2 = 0; t2 < NT; ++t2) {
            cls_sum += s_cls[t2]; box_sum += s_box[t2];
            obj_corr += s_objc[t2]; npos += s_last[t2];
        }
        float obj_l = (0.5f * s_spsum + obj_corr) / (float)ANCH;           // OBJ_W = 1
        float cls_l = (npos > 0) ? cls_sum / (float)(npos * NC) : 0.f;     // CLS_W = 1
        float box_l = (npos > 0) ? box_sum / (float)(npos * 4) * 2.0f : 0.f; // BOX_W = 2
        ws[b * 3 + 0] = box_l;
        ws[b * 3 + 1] = cls_l;
        ws[b * 3 + 2] = obj_l;
    }
}

__global__ void yolo_loss_final(const float* __restrict__ ws, float* __restrict__ out) {
    if (blockIdx.x == 0 && threadIdx.x == 0) {
        float bx = 0.f, cl = 0.f, ob = 0.f;
        for (int b = 0; b < BATCH; ++b) {
            bx += ws[b * 3 + 0];
            cl += ws[b * 3 + 1];
            ob += ws[b * 3 + 2];
        }
        out[0] = bx + cl + ob;
        out[1] = bx;
        out[2] = cl;
        out[3] = ob;
    }
}

extern "C" void kernel_launch(void* const* d_in, const int* in_sizes, int n_in,
                              void* d_out, int out_size, void* d_ws, size_t ws_size,
                              hipStream_t stream) {
    (void)in_sizes; (void)n_in; (void)out_size; (void)ws_size;
    const float* pred = (const float*)d_in[0];
    const float* tbox = (const float*)d_in[1];
    const int*   tcls = (const int*)d_in[2];
    float* ws = (float*)d_ws;   // 32*3 floats of per-sample (box, cls, obj)
    yolo_loss_main<<<BATCH, THREADS, 0, stream>>>(pred, tbox, tcls, ws);
    yolo_loss_final<<<1, 32, 0, stream>>>(ws, (float*)d_out);
}